// KalmanFilter_7310034338112
// MI455X (gfx1250) — compile-verified
//
#include <hip/hip_runtime.h>

typedef float v2f __attribute__((ext_vector_type(2)));
typedef float v8f __attribute__((ext_vector_type(8)));

#define WAVES_PER_BLOCK 4
#define KF_EPS 1e-6f

// Per-wave LDS workspace. 5 padded 16x17 fp32 tiles + small vectors.
struct WaveLds {
  float t[5][16][17];  // t0: F -> T4 ; t1: P -> T1 -> S ; t2: P_pred ; t3: H -> K ; t4: T3=H*P_pred
  float xs[16];        // x_est, 2 batches
  float xp[16];        // x_pred
  float zs[8];         // z
  float yv[8];         // innovation y
  float sInv[2][16];   // S^-1 per batch
};

static __device__ __forceinline__ v8f wmma4(v2f a, v2f b, v8f c) {
  // D = A(16x4) * B(4x16) + C(16x16), fp32
  return __builtin_amdgcn_wmma_f32_16x16x4_f32(false, a, false, b, (short)0, c, false, false);
}

// A operand (16x4 tile at column kbase): VGPR0 = K{kbase,kbase+2}, VGPR1 = K{kbase+1,kbase+3}
static __device__ __forceinline__ v2f ldA(const float (*t)[17], int lane, int kbase) {
  int row = lane & 15, kh = lane >> 4;
  v2f a;
  a.x = t[row][kbase + 2 * kh];
  a.y = t[row][kbase + 2 * kh + 1];
  return a;
}
// B operand (4x16 tile, rows kbase..kbase+3)
static __device__ __forceinline__ v2f ldB(const float (*t)[17], int lane, int kbase) {
  int col = lane & 15, kh = lane >> 4;
  v2f b;
  b.x = t[kbase + 2 * kh][col];
  b.y = t[kbase + 2 * kh + 1][col];
  return b;
}
// B operand = transpose of tile (B[k][n] = t[n][k])
static __device__ __forceinline__ v2f ldBT(const float (*t)[17], int lane, int kbase) {
  int col = lane & 15, kh = lane >> 4;
  v2f b;
  b.x = t[col][kbase + 2 * kh];
  b.y = t[col][kbase + 2 * kh + 1];
  return b;
}
// C/D 16x16 f32 layout: VGPR r = row (r + 8*(lane/16)), col = lane%16
static __device__ __forceinline__ v8f ldC(const float (*t)[17], int lane) {
  int col = lane & 15, kh = lane >> 4;
  v8f c;
#pragma unroll
  for (int r = 0; r < 8; ++r) c[r] = t[r + 8 * kh][col];
  return c;
}
static __device__ __forceinline__ void stD(float (*t)[17], int lane, v8f d) {
  int col = lane & 15, kh = lane >> 4;
#pragma unroll
  for (int r = 0; r < 8; ++r) t[r + 8 * kh][col] = d[r];
}

// 4x4 inverse by adjugate (row-major)
static __device__ __forceinline__ void invert4(const float m[16], float o[16]) {
  float inv[16];
  inv[0]  =  m[5]*m[10]*m[15] - m[5]*m[11]*m[14] - m[9]*m[6]*m[15] + m[9]*m[7]*m[14] + m[13]*m[6]*m[11] - m[13]*m[7]*m[10];
  inv[4]  = -m[4]*m[10]*m[15] + m[4]*m[11]*m[14] + m[8]*m[6]*m[15] - m[8]*m[7]*m[14] - m[12]*m[6]*m[11] + m[12]*m[7]*m[10];
  inv[8]  =  m[4]*m[9]*m[15]  - m[4]*m[11]*m[13] - m[8]*m[5]*m[15] + m[8]*m[7]*m[13] + m[12]*m[5]*m[11] - m[12]*m[7]*m[9];
  inv[12] = -m[4]*m[9]*m[14]  + m[4]*m[10]*m[13] + m[8]*m[5]*m[14] - m[8]*m[6]*m[13] - m[12]*m[5]*m[10] + m[12]*m[6]*m[9];
  inv[1]  = -m[1]*m[10]*m[15] + m[1]*m[11]*m[14] + m[9]*m[2]*m[15] - m[9]*m[3]*m[14] - m[13]*m[2]*m[11] + m[13]*m[3]*m[10];
  inv[5]  =  m[0]*m[10]*m[15] - m[0]*m[11]*m[14] - m[8]*m[2]*m[15] + m[8]*m[3]*m[14] + m[12]*m[2]*m[11] - m[12]*m[3]*m[10];
  inv[9]  = -m[0]*m[9]*m[15]  + m[0]*m[11]*m[13] + m[8]*m[1]*m[15] - m[8]*m[3]*m[13] - m[12]*m[1]*m[11] + m[12]*m[3]*m[9];
  inv[13] =  m[0]*m[9]*m[14]  - m[0]*m[10]*m[13] - m[8]*m[1]*m[14] + m[8]*m[2]*m[13] + m[12]*m[1]*m[10] - m[12]*m[2]*m[9];
  inv[2]  =  m[1]*m[6]*m[15]  - m[1]*m[7]*m[14]  - m[5]*m[2]*m[15] + m[5]*m[3]*m[14] + m[13]*m[2]*m[7]  - m[13]*m[3]*m[6];
  inv[6]  = -m[0]*m[6]*m[15]  + m[0]*m[7]*m[14]  + m[4]*m[2]*m[15] - m[4]*m[3]*m[14] - m[12]*m[2]*m[7]  + m[12]*m[3]*m[6];
  inv[10] =  m[0]*m[5]*m[15]  - m[0]*m[7]*m[13]  - m[4]*m[1]*m[15] + m[4]*m[3]*m[13] + m[12]*m[1]*m[7]  - m[12]*m[3]*m[5];
  inv[14] = -m[0]*m[5]*m[14]  + m[0]*m[6]*m[13]  + m[4]*m[1]*m[14] - m[4]*m[2]*m[13] - m[12]*m[1]*m[6]  + m[12]*m[2]*m[5];
  inv[3]  = -m[1]*m[6]*m[11]  + m[1]*m[7]*m[10]  + m[5]*m[2]*m[11] - m[5]*m[3]*m[10] - m[9]*m[2]*m[7]   + m[9]*m[3]*m[6];
  inv[7]  =  m[0]*m[6]*m[11]  - m[0]*m[7]*m[10]  - m[4]*m[2]*m[11] + m[4]*m[3]*m[10] + m[8]*m[2]*m[7]   - m[8]*m[3]*m[6];
  inv[11] = -m[0]*m[5]*m[11]  + m[0]*m[7]*m[9]   + m[4]*m[1]*m[11] - m[4]*m[3]*m[9]  - m[8]*m[1]*m[7]   + m[8]*m[3]*m[5];
  inv[15] =  m[0]*m[5]*m[10]  - m[0]*m[6]*m[9]   - m[4]*m[1]*m[10] + m[4]*m[2]*m[9]  + m[8]*m[1]*m[6]   - m[8]*m[2]*m[5];
  float det = m[0]*inv[0] + m[1]*inv[4] + m[2]*inv[8] + m[3]*inv[12];
  float rd = 1.0f / det;
#pragma unroll
  for (int i = 0; i < 16; ++i) o[i] = inv[i] * rd;
}

__global__ __launch_bounds__(128) void kalman_wmma_kernel(
    const float* __restrict__ gx, const float* __restrict__ gP,
    const float* __restrict__ gF, const float* __restrict__ gQ,
    const float* __restrict__ gz, const float* __restrict__ gH,
    const float* __restrict__ gR,
    float* __restrict__ ox, float* __restrict__ oP, float* __restrict__ oK,
    int B) {
  __shared__ WaveLds lds[WAVES_PER_BLOCK];
  const int lane = threadIdx.x & 31;
  const int w = threadIdx.x >> 5;
  WaveLds& L = lds[w];

  long gw = (long)blockIdx.x * WAVES_PER_BLOCK + w;
  long b0 = 2 * gw, b1 = b0 + 1;
  if (b0 >= B) b0 = B - 1;  // clamp (deterministic duplicate writes on tail)
  if (b1 >= B) b1 = B - 1;

  const int bi = lane >> 4;   // lower half lanes -> batch0, upper -> batch1 (staging)
  const int li = lane & 15;
  const long bb = bi ? b1 : b0;

  // ---- zero LDS (off-diagonal blocks must be 0) ----
  {
    float* p = (float*)&L;
    const int n = (int)(sizeof(WaveLds) / 4);
    for (int i = lane; i < n; i += 32) p[i] = 0.f;
  }
  __syncthreads();

  // ---- stage inputs block-diagonally ----
  {
    const float4* Fv = (const float4*)(gF + bb * 64);
    const float4* Pv = (const float4*)(gP + bb * 64);
    float4 f4 = Fv[li];
    float4 p4 = Pv[li];
    int e = li * 4, row = (e >> 3) + 8 * bi, c0 = (e & 7) + 8 * bi;
    L.t[0][row][c0] = f4.x; L.t[0][row][c0 + 1] = f4.y; L.t[0][row][c0 + 2] = f4.z; L.t[0][row][c0 + 3] = f4.w;
    L.t[1][row][c0] = p4.x; L.t[1][row][c0 + 1] = p4.y; L.t[1][row][c0 + 2] = p4.z; L.t[1][row][c0 + 3] = p4.w;

    const float2* Hv = (const float2*)(gH + bb * 32);
    float2 h2 = Hv[li];
    int eh = li * 2, rh = (eh >> 3) + 8 * bi, ch = (eh & 7) + 8 * bi;
    L.t[3][rh][ch] = h2.x; L.t[3][rh][ch + 1] = h2.y;

    if (li < 8) L.xs[8 * bi + li] = gx[bb * 8 + li];
    if (li < 4) L.zs[4 * bi + li] = gz[bb * 4 + li];
  }
  __syncthreads();

  // ---- x_pred = F x (VALU, 16 lanes = 2 batches x 8 rows) ----
  if (lane < 16) {
    int pb = lane >> 3, r = lane & 7;
    float s = 0.f;
#pragma unroll
    for (int c = 0; c < 8; ++c) s += L.t[0][r + 8 * pb][c + 8 * pb] * L.xs[8 * pb + c];
    L.xp[lane] = s;
  }
  __syncthreads();
  // ---- y = z - H x_pred (8 lanes) ----
  if (lane < 8) {
    int pb = lane >> 2, r = lane & 3;
    float s = 0.f;
#pragma unroll
    for (int c = 0; c < 8; ++c) s += L.t[3][r + 8 * pb][c + 8 * pb] * L.xp[8 * pb + c];
    L.yv[lane] = L.zs[lane] - s;
  }
  __syncthreads();

  v8f acc;
  v8f cz = {};

  // ---- step1: T1 = F * P  (t0 x t1 -> t1) ----
  acc = wmma4(ldA(L.t[0], lane, 0), ldB(L.t[1], lane, 0), cz);
  acc = wmma4(ldA(L.t[0], lane, 4), ldB(L.t[1], lane, 4), acc);
  stD(L.t[1], lane, acc);
  __syncthreads();

  // ---- step2: P_pred = T1 * F^T + Q  (Q streamed into C operand) ----
  v8f cq = {};
  {
    int col = lane & 15, kh = lane >> 4;
    if ((col >> 3) == kh) {
      int cc = col & 7;
      long b = kh ? b1 : b0;
#pragma unroll
      for (int r = 0; r < 8; ++r) cq[r] = gQ[b * 64 + r * 8 + cc];
    }
  }
  acc = wmma4(ldA(L.t[1], lane, 0), ldBT(L.t[0], lane, 0), cq);
  acc = wmma4(ldA(L.t[1], lane, 4), ldBT(L.t[0], lane, 4), acc);
  stD(L.t[2], lane, acc);
  __syncthreads();

  // ---- step3: T3 = H * P_pred ----
  acc = wmma4(ldA(L.t[3], lane, 0), ldB(L.t[2], lane, 0), cz);
  acc = wmma4(ldA(L.t[3], lane, 4), ldB(L.t[2], lane, 4), acc);
  stD(L.t[4], lane, acc);
  __syncthreads();

  // ---- step4: S = T3 * H^T + (R + eps*I) ----
  v8f cr = {};
  {
    int col = lane & 15, kh = lane >> 4;
    if ((col >> 3) == kh) {
      int cc = col & 7;
      if (cc < 4) {
        long b = kh ? b1 : b0;
#pragma unroll
        for (int r = 0; r < 4; ++r)
          cr[r] = gR[b * 16 + r * 4 + cc] + (r == cc ? KF_EPS : 0.f);
      }
    }
  }
  acc = wmma4(ldA(L.t[4], lane, 0), ldBT(L.t[3], lane, 0), cr);
  acc = wmma4(ldA(L.t[4], lane, 4), ldBT(L.t[3], lane, 4), acc);
  stD(L.t[1], lane, acc);
  __syncthreads();

  // ---- invert S (one lane per batch) ----
  if ((lane & 15) == 0) {
    int pb = lane >> 4;
    float m[16], o[16];
#pragma unroll
    for (int i = 0; i < 16; ++i) m[i] = L.t[1][8 * pb + (i >> 2)][8 * pb + (i & 3)];
    invert4(m, o);
#pragma unroll
    for (int i = 0; i < 16; ++i) L.sInv[pb][i] = o[i];
  }
  __syncthreads();

  // ---- step5: T4 = P_pred * H^T ----
  acc = wmma4(ldA(L.t[2], lane, 0), ldBT(L.t[3], lane, 0), cz);
  acc = wmma4(ldA(L.t[2], lane, 4), ldBT(L.t[3], lane, 4), acc);
  stD(L.t[0], lane, acc);
  __syncthreads();

  // ---- step6: K = T4 * S^-1  (K-dim = 4, single WMMA; batch1 K-indices compacted) ----
  v2f a6, b6;
  {
    int row = lane & 15, kh = lane >> 4;
    int kofs = (row >= 8) ? 8 : 0;   // batch1 T4 columns live at 8..11
    a6.x = L.t[0][row][kofs + 2 * kh];
    a6.y = L.t[0][row][kofs + 2 * kh + 1];
    int col = lane & 15, cb = col >> 3, cc = col & 7, k = 2 * kh;
    b6.x = (cc < 4) ? L.sInv[cb][k * 4 + cc] : 0.f;
    b6.y = (cc < 4) ? L.sInv[cb][(k + 1) * 4 + cc] : 0.f;
  }
  acc = wmma4(a6, b6, cz);
  stD(L.t[3], lane, acc);
  __syncthreads();

  // ---- x_new = x_pred + K y ; write x_new and K ----
  if (lane < 16) {
    int pb = lane >> 3, r = lane & 7;
    long b = pb ? b1 : b0;
    float s = L.xp[lane];
#pragma unroll
    for (int j = 0; j < 4; ++j) s += L.t[3][r + 8 * pb][j + 8 * pb] * L.yv[4 * pb + j];
    ox[b * 8 + r] = s;
  }
#pragma unroll
  for (int e = 0; e < 2; ++e) {
    int f = lane * 2 + e;
    int pb = f >> 5, idx = f & 31;
    long b = pb ? b1 : b0;
    oK[b * 32 + idx] = L.t[3][(idx >> 2) + 8 * pb][(idx & 3) + 8 * pb];
  }

  // ---- step7: P_new = (-K) * T3 + P_pred  (single WMMA), stream D to HBM ----
  v2f a7, b7;
  {
    int row = lane & 15, kh = lane >> 4;
    int kofs = (row >= 8) ? 8 : 0;   // batch1 K columns live at 8..11
    a7.x = -L.t[3][row][kofs + 2 * kh];
    a7.y = -L.t[3][row][kofs + 2 * kh + 1];
    int col = lane & 15, rofs = (col >= 8) ? 8 : 0;  // batch1 T3 rows live at 8..11
    b7.x = L.t[4][2 * kh + rofs][col];
    b7.y = L.t[4][2 * kh + 1 + rofs][col];
  }
  v8f c7 = ldC(L.t[2], lane);
  v8f d7 = wmma4(a7, b7, c7);
  {
    int col = lane & 15, kh = lane >> 4;
    if ((col >> 3) == kh) {        // diagonal block only
      long b = kh ? b1 : b0;
      int cc = col & 7;
#pragma unroll
      for (int r = 0; r < 8; ++r) oP[b * 64 + r * 8 + cc] = d7[r];
    }
  }
}

extern "C" void kernel_launch(void* const* d_in, const int* in_sizes, int n_in,
                              void* d_out, int out_size, void* d_ws, size_t ws_size,
                              hipStream_t stream) {
  const int B = in_sizes[0] / 8;  // x_est is [B,8,1]
  const float* gx = (const float*)d_in[0];
  const float* gP = (const float*)d_in[1];
  const float* gF = (const float*)d_in[2];
  const float* gQ = (const float*)d_in[3];
  const float* gz = (const float*)d_in[4];
  const float* gH = (const float*)d_in[5];
  const float* gR = (const float*)d_in[6];

  float* ox = (float*)d_out;                 // [B,8]
  float* oP = ox + (size_t)B * 8;            // [B,8,8]
  float* oK = oP + (size_t)B * 64;           // [B,8,4]

  const int batchesPerBlock = 2 * WAVES_PER_BLOCK;  // 2 per wave
  dim3 block(32 * WAVES_PER_BLOCK);
  dim3 grid((B + batchesPerBlock - 1) / batchesPerBlock);
  kalman_wmma_kernel<<<grid, block, 0, stream>>>(gx, gP, gF, gQ, gz, gH, gR, ox, oP, oK, B);
}